// Image_Model_30520037605625
// MI455X (gfx1250) — compile-verified
//
#include <hip/hip_runtime.h>

#define NS 8192
#define NG 4096
#define NL 512
#define CHUNKS 32                 // row chunks for the gene pass
#define ROWS_PER_CHUNK (NS / CHUNKS)   // 256
#define GENE_GROUPS (NG / 16)          // 256
#define GENE_BLOCKS ((GENE_GROUPS * CHUNKS) / 8)   // 1024 blocks of 8 waves

typedef __attribute__((ext_vector_type(2))) float v2f;
typedef __attribute__((ext_vector_type(8))) float v8f;

// ---------------------------------------------------------------------------
// Kernel 1: per-gene stats (5 column sums over spots) via V_WMMA_F32_16X16X4_F32
// with A = ones (exact f32 column reduction), plus MSE partials.
// Wave w owns genes [16g, 16g+16) and rows [256c, 256c+256).
// B layout per step: lanes 0-15 -> (row r0+2*0 / r0+1, col=lane), halves give
// 4 rows per WMMA (K=4). A all-ones => D[m][n] = column-n sum, any K order.
// ---------------------------------------------------------------------------
__global__ void __launch_bounds__(256) gene_stats_kernel(
    const float* __restrict__ feat, const float* __restrict__ preds,
    const float* __restrict__ nuclei, float* __restrict__ geneP,
    float* __restrict__ msePart)
{
    const int wid    = threadIdx.x >> 5;
    const int lane   = threadIdx.x & 31;
    const int waveId = blockIdx.x * 8 + wid;       // 0..8191
    const int geneGroup = waveId & (GENE_GROUPS - 1);
    const int chunk     = waveId >> 8;             // 0..31
    const int col  = lane & 15;
    const int half = lane >> 4;
    const int gbase = geneGroup * 16;
    const int rbase = chunk * ROWS_PER_CHUNK;

    const v2f Aones = {1.0f, 1.0f};
    v8f Cf = {}, Co = {}, Cff = {}, Coo = {}, Cfo = {};
    float mse = 0.0f;

    const float* fp = feat  + (size_t)(rbase + 2 * half) * NG + gbase + col;
    const float* pp = preds + (size_t)(rbase + 2 * half) * NG + gbase + col;
    const float* np = nuclei + rbase + 2 * half;

    #pragma unroll 2
    for (int s = 0; s < ROWS_PER_CHUNK / 4; ++s) {
        float f0 = __builtin_nontemporal_load(fp);
        float f1 = __builtin_nontemporal_load(fp + NG);
        float p0 = __builtin_nontemporal_load(pp);
        float p1 = __builtin_nontemporal_load(pp + NG);
        float n0 = np[0];
        float n1 = np[1];
        fp += 4 * NG; pp += 4 * NG; np += 4;

        float o0 = p0 * n0, o1 = p1 * n1;          // out = preds * nuclei
        v2f Bf  = {f0, f1};
        v2f Bo  = {o0, o1};
        v2f Bff = {f0 * f0, f1 * f1};
        v2f Boo = {o0 * o0, o1 * o1};
        v2f Bfo = {f0 * o0, f1 * o1};

        Cf  = __builtin_amdgcn_wmma_f32_16x16x4_f32(false, Aones, false, Bf,  (short)0, Cf,  false, false);
        Co  = __builtin_amdgcn_wmma_f32_16x16x4_f32(false, Aones, false, Bo,  (short)0, Co,  false, false);
        Cff = __builtin_amdgcn_wmma_f32_16x16x4_f32(false, Aones, false, Bff, (short)0, Cff, false, false);
        Coo = __builtin_amdgcn_wmma_f32_16x16x4_f32(false, Aones, false, Boo, (short)0, Coo, false, false);
        Cfo = __builtin_amdgcn_wmma_f32_16x16x4_f32(false, Aones, false, Bfo, (short)0, Cfo, false, false);

        float d0 = o0 - f0, d1 = o1 - f1;
        mse = fmaf(d0, d0, mse);
        mse = fmaf(d1, d1, mse);
    }

    // All D rows are identical (A == ones): every lane's C[0] is the column
    // sum for col = lane&15. Lanes 0-15 publish partials.
    if (half == 0) {
        float* base = geneP + (size_t)chunk * NG + gbase + col;
        base[0 * CHUNKS * NG] = Cf[0];
        base[1 * CHUNKS * NG] = Co[0];
        base[2 * CHUNKS * NG] = Cff[0];
        base[3 * CHUNKS * NG] = Coo[0];
        base[4 * CHUNKS * NG] = Cfo[0];
    }

    __shared__ float red[256];
    red[threadIdx.x] = mse;
    __syncthreads();
    for (int off = 128; off > 0; off >>= 1) {
        if (threadIdx.x < off) red[threadIdx.x] += red[threadIdx.x + off];
        __syncthreads();
    }
    if (threadIdx.x == 0) msePart[blockIdx.x] = red[0];
}

// ---------------------------------------------------------------------------
// Kernel 2: per-row Pearson over the 512-dim latents. One wave per spot,
// 16 floats/lane via float4 loads, shuffle tree reduction.
// ---------------------------------------------------------------------------
__global__ void __launch_bounds__(256) latent_kernel(
    const float* __restrict__ a, const float* __restrict__ b,
    float* __restrict__ rlat)
{
    const int wid  = threadIdx.x >> 5;
    const int lane = threadIdx.x & 31;
    const int spot = blockIdx.x * 8 + wid;        // grid 1024 -> 8192 spots

    const float4* pa = (const float4*)(a + (size_t)spot * NL + lane * 16);
    const float4* pb = (const float4*)(b + (size_t)spot * NL + lane * 16);

    float sa = 0.f, sb = 0.f, saa = 0.f, sbb = 0.f, sab = 0.f;
    #pragma unroll
    for (int i = 0; i < 4; ++i) {
        float4 va = pa[i];
        float4 vb = pb[i];
        sa += va.x; sb += vb.x; saa = fmaf(va.x, va.x, saa); sbb = fmaf(vb.x, vb.x, sbb); sab = fmaf(va.x, vb.x, sab);
        sa += va.y; sb += vb.y; saa = fmaf(va.y, va.y, saa); sbb = fmaf(vb.y, vb.y, sbb); sab = fmaf(va.y, vb.y, sab);
        sa += va.z; sb += vb.z; saa = fmaf(va.z, va.z, saa); sbb = fmaf(vb.z, vb.z, sbb); sab = fmaf(va.z, vb.z, sab);
        sa += va.w; sb += vb.w; saa = fmaf(va.w, va.w, saa); sbb = fmaf(vb.w, vb.w, sbb); sab = fmaf(va.w, vb.w, sab);
    }
    #pragma unroll
    for (int off = 16; off > 0; off >>= 1) {
        sa  += __shfl_xor(sa,  off, 32);
        sb  += __shfl_xor(sb,  off, 32);
        saa += __shfl_xor(saa, off, 32);
        sbb += __shfl_xor(sbb, off, 32);
        sab += __shfl_xor(sab, off, 32);
    }
    if (lane == 0) {
        const float inv = 1.0f / (float)NL;
        float mu1 = sa * inv, mu2 = sb * inv;
        float s12 = sab * inv - mu1 * mu2;
        float s1  = sqrtf(saa * inv - mu1 * mu1);
        float s2  = sqrtf(sbb * inv - mu2 * mu2);
        float r   = s12 / (s1 * s2);
        rlat[spot] = (r != r) ? 0.0f : r;   // NaN -> 0, like the reference
    }
}

// ---------------------------------------------------------------------------
// Kernel 3: deterministic finalize (single block).
// ---------------------------------------------------------------------------
__global__ void __launch_bounds__(256) finalize_kernel(
    const float* __restrict__ geneP, const float* __restrict__ msePart,
    const float* __restrict__ rlat, float* __restrict__ out)
{
    const int tid = threadIdx.x;

    float mseSum = 0.0f;
    for (int i = tid; i < GENE_BLOCKS; i += 256) mseSum += msePart[i];

    float rlSum = 0.0f;
    for (int i = tid; i < NS; i += 256) rlSum += rlat[i];

    float rgSum = 0.0f;
    for (int g = tid; g < NG; g += 256) {
        float S0 = 0.f, S1 = 0.f, S2 = 0.f, S3 = 0.f, S4 = 0.f;
        for (int c = 0; c < CHUNKS; ++c) {
            const float* p = geneP + (size_t)c * NG + g;
            S0 += p[0 * CHUNKS * NG];
            S1 += p[1 * CHUNKS * NG];
            S2 += p[2 * CHUNKS * NG];
            S3 += p[3 * CHUNKS * NG];
            S4 += p[4 * CHUNKS * NG];
        }
        const float inv = 1.0f / (float)NS;
        float mu1 = S0 * inv, mu2 = S1 * inv;
        float s12 = S4 * inv - mu1 * mu2;
        float s1  = sqrtf(S2 * inv - mu1 * mu1);
        float s2  = sqrtf(S3 * inv - mu2 * mu2);
        float r   = s12 / (s1 * s2);
        rgSum += (r != r) ? 0.0f : r;
    }

    __shared__ float rA[256], rB[256], rC[256];
    rA[tid] = mseSum; rB[tid] = rlSum; rC[tid] = rgSum;
    __syncthreads();
    for (int off = 128; off > 0; off >>= 1) {
        if (tid < off) {
            rA[tid] += rA[tid + off];
            rB[tid] += rB[tid + off];
            rC[tid] += rC[tid + off];
        }
        __syncthreads();
    }
    if (tid == 0) {
        out[0] = 1.0f - rB[0] / (float)NS;                 // latent_simloss
        out[1] = rA[0] / ((float)NS * (float)NG);          // mse_loss
        out[2] = 1.0f - rC[0] / (float)NG;                 // pcc_loss
    }
}

extern "C" void kernel_launch(void* const* d_in, const int* in_sizes, int n_in,
                              void* d_out, int out_size, void* d_ws, size_t ws_size,
                              hipStream_t stream) {
    (void)in_sizes; (void)n_in; (void)out_size; (void)ws_size;
    const float* feat = (const float*)d_in[0];   // spot_features [8192,4096]
    const float* prds = (const float*)d_in[1];   // preds         [8192,4096]
    const float* lat  = (const float*)d_in[2];   // spot_latent   [8192,512]
    const float* img  = (const float*)d_in[3];   // image_embeds  [8192,512]
    const float* nuc  = (const float*)d_in[4];   // nuclei        [8192,1]

    float* geneP   = (float*)d_ws;                        // 5*CHUNKS*NG floats (2.5 MB)
    float* msePart = geneP + 5 * CHUNKS * NG;             // GENE_BLOCKS floats
    float* rlat    = msePart + GENE_BLOCKS;               // NS floats

    gene_stats_kernel<<<GENE_BLOCKS, 256, 0, stream>>>(feat, prds, nuc, geneP, msePart);
    latent_kernel<<<NS / 8, 256, 0, stream>>>(lat, img, rlat);
    finalize_kernel<<<1, 256, 0, stream>>>(geneP, msePart, rlat, (float*)d_out);
}